// WeightedBusePenaltyLoss_30794915512726
// MI455X (gfx1250) — compile-verified
//
#include <hip/hip_runtime.h>
#include <math.h>

// WMMA register types for V_WMMA_F32_16X16X4_F32 (wave32):
//   A: 16x4 f32  -> 2 VGPRs ; C/D: 16x16 f32 -> 8 VGPRs
typedef float v2f __attribute__((ext_vector_type(2)));
typedef float v8f __attribute__((ext_vector_type(8)));

#define DIMS    256
#define WAVES   4            // waves (tiles) per 128-thread block
#define KCHUNK  64           // k-slice staged in LDS per step
#define LSTR    68           // padded floats per staged row (68%64==4 -> no bank conflicts)
#define PENALTY 256.0f       // MULT * D ('dim' penalty option)
#define LN2_F   0.69314718055994530942f
#define EPS_F   1e-6f

// ---------------------------------------------------------------------------
// Stage 1: each wave computes one 16-row tile.
//   s1[m] = sum_k pred[m,k]^2               via diag( P * P^T )
//   s2[m] = sum_k (proto[m,k]-pred[m,k])^2  via diag( D * D^T )
//
// Data movement: pred/proto k-slices are DMA'd into LDS with
// GLOBAL_LOAD_ASYNC_TO_LDS_B128 (coalesced: 512B / 4 cache lines per instr),
// then the WMMA operand pattern (lane L -> row L%16, k-block by lane half)
// is read from LDS conflict-free thanks to the 68-float row stride.
// ---------------------------------------------------------------------------
__global__ __launch_bounds__(128) void buse_loss_stage1(
    const float* __restrict__ pred,      // [N, 256]
    const int*   __restrict__ target,    // [N], values in [0, 81)
    const float* __restrict__ polars,    // [81, 256] unit rows
    const float* __restrict__ cw,        // [81]
    float*       __restrict__ partials)  // [gridDim.x]
{
    __shared__ __attribute__((aligned(16))) float stage[WAVES][2][16][LSTR];
    __shared__ float sh1[WAVES][16][17];
    __shared__ float sh2[WAVES][16][17];
    __shared__ float waveSum[WAVES];

    const int lane      = threadIdx.x & 31;
    const int wave      = threadIdx.x >> 5;
    const int tile      = blockIdx.x * WAVES + wave;
    const int rowInTile = lane & 15;
    const int half      = lane >> 4;            // K-block selector for WMMA A/B
    const int row       = tile * 16 + rowInTile;

    // Double-gathered class weight, faithful to the reference:
    //   w = class_weights[target[target[i]]]
    const int   t = target[row];
    const float w = cw[target[t]];

    // Async-copy geometry: instruction i covers rows {2i, 2i+1} of the tile;
    // lanes 0-15 handle the first row's 256B slice, lanes 16-31 the second.
    const int sub   = lane >> 4;                 // which of the two rows
    const int off16 = (lane & 15) * 16;          // byte offset within 256B slice

    v8f c1 = {};  // accumulates P * P^T
    v8f c2 = {};  // accumulates (Q-P) * (Q-P)^T

    const float2* __restrict__ pslice = (const float2*)&stage[wave][0][rowInTile][0];
    const float2* __restrict__ qslice = (const float2*)&stage[wave][1][rowInTile][0];

    for (int c = 0; c < DIMS / KCHUNK; ++c) {
        const int k0 = c * KCHUNK;

        // ---- stage this chunk's pred + gathered-proto slices into LDS ----
        #pragma unroll
        for (int i = 0; i < 8; ++i) {
            const int r = 2 * i + sub;
            // pred[tile*16 + r][k0 .. k0+63]
            const char* gp = (const char*)(pred + (size_t)(tile * 16 + r) * DIMS + k0) + off16;
            unsigned    lp = (unsigned)(size_t)(&stage[wave][0][r][0]) + (unsigned)off16;
            asm volatile("global_load_async_to_lds_b128 %0, %1, off"
                         :: "v"(lp), "v"(gp) : "memory");
            // polars[target[row r]][k0 .. k0+63]
            const int   tr = __shfl(t, r, 32);
            const char* gq = (const char*)(polars + (size_t)tr * DIMS + k0) + off16;
            unsigned    lq = (unsigned)(size_t)(&stage[wave][1][r][0]) + (unsigned)off16;
            asm volatile("global_load_async_to_lds_b128 %0, %1, off"
                         :: "v"(lq), "v"(gq) : "memory");
        }
        asm volatile("s_wait_asynccnt 0x0" ::: "memory");

        // ---- consume: 16 WMMA k-steps over the staged slice ----
        #pragma unroll 4
        for (int kl = 0; kl < KCHUNK; kl += 4) {
            const int idx2 = (kl >> 1) + half;   // float2 index: k = kl + 2*half
            const float2 p = pslice[idx2];       // ds_load_b64, conflict-free
            const float2 q = qslice[idx2];
            v2f a; a.x = p.x;       a.y = p.y;
            v2f d; d.x = q.x - p.x; d.y = q.y - p.y;
            // 8-arg form: (neg_a, A, neg_b, B, c_mod, C, reuse_a, reuse_b)
            c1 = __builtin_amdgcn_wmma_f32_16x16x4_f32(false, a, false, a,
                                                       (short)0, c1, false, false);
            c2 = __builtin_amdgcn_wmma_f32_16x16x4_f32(false, d, false, d,
                                                       (short)0, c2, false, false);
        }
    }

    // Scatter the 16x16 accumulators to LDS; C/D layout:
    // lanes 0-15: M = v, N = lane ; lanes 16-31: M = v+8, N = lane-16.
    const int mBase = half * 8;
    #pragma unroll
    for (int v = 0; v < 8; ++v) {
        sh1[wave][mBase + v][rowInTile] = c1[v];
        sh2[wave][mBase + v][rowInTile] = c2[v];
    }
    __syncthreads();

    float loss = 0.0f;
    if (lane < 16) {
        const float nsq_diff = sh2[wave][rowInTile][rowInTile]; // ||q-p||^2
        const float nsq_pred = sh1[wave][rowInTile][rowInTile]; // ||p||^2
        const float diff_log  = __logf(nsq_diff);               // 2*log||q-p||
        const float proto_log = (1.0f + PENALTY) *
                                __logf(1.0f - nsq_pred + EPS_F);
        loss = (diff_log - proto_log + PENALTY * LN2_F) * w;
    }

    // wave32 tree reduction
    #pragma unroll
    for (int off = 16; off > 0; off >>= 1)
        loss += __shfl_down(loss, off, 32);
    if (lane == 0) waveSum[wave] = loss;
    __syncthreads();

    if (threadIdx.x == 0) {
        float s = 0.0f;
        #pragma unroll
        for (int i = 0; i < WAVES; ++i) s += waveSum[i];
        partials[blockIdx.x] = s;
    }
}

// ---------------------------------------------------------------------------
// Stage 2: single block folds the per-block partials and applies 1/N.
// ---------------------------------------------------------------------------
__global__ __launch_bounds__(256) void buse_loss_stage2(
    const float* __restrict__ partials, int n,
    float* __restrict__ out, float invN)
{
    __shared__ float ws[8];
    float s = 0.0f;
    for (int i = threadIdx.x; i < n; i += 256) s += partials[i];
    #pragma unroll
    for (int off = 16; off > 0; off >>= 1)
        s += __shfl_down(s, off, 32);
    const int lane = threadIdx.x & 31, wave = threadIdx.x >> 5;
    if (lane == 0) ws[wave] = s;
    __syncthreads();
    if (threadIdx.x == 0) {
        float tot = 0.0f;
        #pragma unroll
        for (int i = 0; i < 8; ++i) tot += ws[i];
        out[0] = tot * invN;
    }
}

extern "C" void kernel_launch(void* const* d_in, const int* in_sizes, int n_in,
                              void* d_out, int out_size, void* d_ws, size_t ws_size,
                              hipStream_t stream) {
    const float* pred   = (const float*)d_in[0];
    const int*   target = (const int*)  d_in[1];
    const float* polars = (const float*)d_in[2];
    const float* cw     = (const float*)d_in[3];
    float* out      = (float*)d_out;
    float* partials = (float*)d_ws;

    const int N      = in_sizes[1];        // 131072
    const int tiles  = N / 16;             // 8192
    const int blocks = tiles / WAVES;      // 2048

    buse_loss_stage1<<<blocks, 128, 0, stream>>>(pred, target, polars, cw, partials);
    buse_loss_stage2<<<1, 256, 0, stream>>>(partials, blocks, out, 1.0f / (float)N);
}